// PositionalEmbedding_37357625541234
// MI455X (gfx1250) — compile-verified
//
#include <hip/hip_runtime.h>

// PositionalEmbedding expansion via one-hot WMMA gather (gfx1250 / CDNA5).
//
// out[(i*B + b)*256 + d] = (i < total[b]) ? encoding[pos(i,b)][d] : 0
// with pos in [0,7] because all durations < 8. Implemented as
//   OutTile[16x16] = OneHot[16x4] x Enc[4x16]  (two K=4 chunks, chained C)
// using V_WMMA_F32_16X16X4_F32 — exact in fp32 (x1.0, single-nonzero sums).

typedef __attribute__((ext_vector_type(2))) float v2f;
typedef __attribute__((ext_vector_type(8))) float v8f;

#define S_TOK 1024   // tokens per sequence
#define NB    32     // batch
#define DIMV  256    // embedding dim
#define NCT   16     // column tiles: 256 / 16

// ---------------------------------------------------------------------------
// Kernel 1: per-batch inclusive cumsum of durations.
// durations: (S, B) int32 row-major.  cum: (B, S) int32 (transposed so the
// per-batch binary search in kernel 2 walks a contiguous 4KB column).
// One wave, lane b scans batch b; loads are coalesced across lanes per step.
// ---------------------------------------------------------------------------
__global__ void pe_cumsum_kernel(const int* __restrict__ dur,
                                 int* __restrict__ cum) {
    int b = threadIdx.x;
    if (b >= NB) return;
    int run = 0;
    for (int t = 0; t < S_TOK; ++t) {
        run += dur[t * NB + b];
        cum[b * S_TOK + t] = run;
    }
}

// ---------------------------------------------------------------------------
// Kernel 2: expand via WMMA one-hot matmul.
// Each wave owns 16 consecutive output rows (row = i*32 + b) per tile and
// loops over 16 column tiles of the 256-wide embedding.
// ---------------------------------------------------------------------------
__global__ void __launch_bounds__(256)
pe_expand_kernel(const int* __restrict__ dur,     // (S, B)
                 const float* __restrict__ enc,   // (L, 256), rows 0..7 used
                 const int* __restrict__ cum,     // (B, S) inclusive cumsum
                 float* __restrict__ out,         // (T*B, 256)
                 int numTiles) {                  // T*B / 16
    const int lane  = threadIdx.x & 31;
    const int wib   = threadIdx.x >> 5;                 // wave in block
    const int wpb   = blockDim.x >> 5;
    const int wave  = blockIdx.x * wpb + wib;
    const int nwave = gridDim.x * wpb;

    const int n     = lane & 15;          // column within tile (B/D lanes 0-15)
    const int khalf = (lane >> 4) << 1;   // 0 for lanes 0-15, 2 for lanes 16-31
    const int mOff  = (lane >> 4) << 3;   // C/D: rows v (lo half) / v+8 (hi half)

    // Preload WMMA-B fragments for encoding rows 0..7, all 16 column tiles.
    // B 4x16 fp32 layout: VGPR0 = row K=(base+khalf), VGPR1 = row K+1,
    // lane n selects the column. 64 VGPRs, loaded once per wave (L2-resident).
    v2f bmat[NCT][2];
#pragma unroll
    for (int c = 0; c < NCT; ++c) {
#pragma unroll
        for (int kk = 0; kk < 2; ++kk) {
            const int k0 = kk * 4 + khalf;
            bmat[c][kk].x = enc[(size_t)k0 * DIMV + c * 16 + n];
            bmat[c][kk].y = enc[(size_t)(k0 + 1) * DIMV + c * 16 + n];
        }
    }

    for (int tile = wave; tile < numTiles; tile += nwave) {
        // Each lane resolves the frame->position mapping for row (lane&15)
        // of this tile (duplicated across lane halves; both halves need it
        // to build their K-slice of the one-hot A fragment).
        const int r = tile * 16 + n;       // global row id in [0, T*B)
        const int i = r >> 5;              // frame index  (B == 32)
        const int b = r & 31;              // batch index
        const int* cb = cum + b * S_TOK;
        const int total = cb[S_TOK - 1];

        int pos = -1;                      // -1 => invalid frame => zero row
        if (i < total) {
            // searchsorted(cum, i, side='right'): first t with cum[t] > i.
            int lo = 0, hi = S_TOK;
            while (lo < hi) {
                const int mid = (lo + hi) >> 1;
                if (cb[mid] > i) hi = mid; else lo = mid + 1;
            }
            const int start = cb[lo] - dur[lo * NB + b];
            pos = i - start;               // in [0, 7): durations < MAX_DUR=8
        }

        // One-hot A fragments (A 16x4 fp32 layout: VGPR0 = K {0|2}, VGPR1 = K+1).
        v2f a0, a1;
        a0.x = (pos == khalf)     ? 1.0f : 0.0f;
        a0.y = (pos == khalf + 1) ? 1.0f : 0.0f;
        a1.x = (pos == khalf + 4) ? 1.0f : 0.0f;
        a1.y = (pos == khalf + 5) ? 1.0f : 0.0f;

        float* orow = out + (size_t)tile * 16 * DIMV;
#pragma unroll
        for (int c = 0; c < NCT; ++c) {
            v8f acc = {};
            // D = A1*B1 + (A0*B0 + 0): chain C through both K=4 chunks.
            acc = __builtin_amdgcn_wmma_f32_16x16x4_f32(
                false, a0, false, bmat[c][0], (short)0, acc, false, false);
            acc = __builtin_amdgcn_wmma_f32_16x16x4_f32(
                false, a1, false, bmat[c][1], (short)0, acc, false, false);
#pragma unroll
            for (int v = 0; v < 8; ++v) {
                orow[(size_t)(v + mOff) * DIMV + c * 16 + n] = acc[v];
            }
        }
    }
}

// ---------------------------------------------------------------------------
extern "C" void kernel_launch(void* const* d_in, const int* in_sizes, int n_in,
                              void* d_out, int out_size, void* d_ws, size_t ws_size,
                              hipStream_t stream) {
    const int*   dur = (const int*)d_in[0];    // (1024, 32) int32
    const float* enc = (const float*)d_in[1];  // (1000, 256) fp32
    float*       out = (float*)d_out;          // (T, 32, 256) fp32
    int*         cum = (int*)d_ws;             // (32, 1024) int32 = 128 KB

    if (out_size <= 0) return;
    const int T = out_size / (NB * DIMV);
    const int numTiles = (T * NB) / 16;        // B=32 => always exact

    hipLaunchKernelGGL(pe_cumsum_kernel, dim3(1), dim3(32), 0, stream, dur, cum);

    int blocks = (numTiles + 7) / 8;           // 8 waves/block, 1 tile per wave pass
    if (blocks > 1024) blocks = 1024;
    if (blocks < 1)    blocks = 1;
    hipLaunchKernelGGL(pe_expand_kernel, dim3(blocks), dim3(256), 0, stream,
                       dur, enc, cum, out, numTiles);
}